// Modified_grouped_shift_gcn_23261542875775
// MI455X (gfx1250) — compile-verified
//
#include <hip/hip_runtime.h>

typedef float v2f __attribute__((ext_vector_type(2)));
typedef float v8f __attribute__((ext_vector_type(8)));

#define VN   25
#define TDIM 300
#define NB   64
#define NT   (NB * TDIM)      // 19200 rows
#define ROWS (NT * VN)        // 480000 GEMM rows
#define COLS (VN * 64)        // 1600 columns
#define LDA  68               // LDS stride (floats): conflict-free, 8B aligned
#define BN_EPS 1e-5f

// ---------------- K0: mask = tanh(fm) + 1 ----------------
__global__ void k_mask(const float* __restrict__ fm, float* __restrict__ mask) {
    int j = blockIdx.x * blockDim.x + threadIdx.x;
    if (j < COLS) mask[j] = tanhf(fm[j]) + 1.0f;
}

// ---------------- K1: gather(shift_in) * mask -> WMMA f32 GEMM -> Y ----------------
// Block = 256 threads = 8 waves; each wave owns a 16-row x 64-col output tile.
__global__ __launch_bounds__(256) void k_gemm(
    const float* __restrict__ x0, const float* __restrict__ W,
    const float* __restrict__ bias, const int* __restrict__ shift_in,
    const float* __restrict__ mask, float* __restrict__ Y)
{
    __shared__ float ldsA[128 * LDA];   // 128 rows x 64 cols (padded)
    __shared__ float ldsW[64 * LDA];    // W transposed: ldsW[d*LDA + k] = W[k][d]

    const int tid      = threadIdx.x;
    const int rowBlock = blockIdx.x * 128;

    // Stage W^T (coalesced global reads).
    #pragma unroll
    for (int i = 0; i < 16; ++i) {
        int e = tid + 256 * i;              // 4096 elements
        int d = e & 63, k = e >> 6;
        ldsW[d * LDA + k] = W[k * 64 + d];
    }
    // Stage gathered + masked A tile: A[row][c] = x0[n, c', t, v'] * mask[v*64+c]
    #pragma unroll
    for (int i = 0; i < 32; ++i) {
        int e   = tid + 256 * i;            // 8192 elements
        int row = e >> 6, c = e & 63;
        int r   = rowBlock + row;
        int nt  = r / VN;  int v = r - nt * VN;
        int n   = nt / TDIM; int t = nt - n * TDIM;
        int j   = v * 64 + c;
        int si  = shift_in[j];              // flattened (v', c'), C=64 pow2
        int vs  = si >> 6, cs = si & 63;
        float a = x0[((n * 64 + cs) * TDIM + t) * VN + vs];
        ldsA[row * LDA + c] = a * mask[j];
    }
    __syncthreads();

    const int wv    = tid >> 5;            // wave id 0..7
    const int lane  = tid & 31;
    const int m     = lane & 15;           // M (A) / N (B,C,D) sub-index
    const int khalf = (lane >> 4) << 1;    // K offset 0 or 2 within K=4 step
    const int rowW  = wv * 16;

    // Output row base per accumulator VGPR g: M = g + 8*(lane>=16)
    int ybase[8];
    #pragma unroll
    for (int g = 0; g < 8; ++g) {
        int M  = g + ((lane >> 4) << 3);
        int r  = rowBlock + rowW + M;
        int nt = r / VN; int v = r - nt * VN;
        ybase[g] = nt * COLS + v * 64;
    }

    #pragma unroll
    for (int d0 = 0; d0 < 64; d0 += 16) {
        v8f acc = {};
        #pragma unroll
        for (int kk = 0; kk < 16; ++kk) {
            // A 16x4 f32 frag: lane m holds A[m][4kk+khalf .. +1]  (ds_load_b64)
            v2f a = *(const v2f*)&ldsA[(rowW + m) * LDA + 4 * kk + khalf];
            // B 4x16 f32 frag: lane m holds W[4kk+khalf .. +1][d0+m] from W^T
            v2f b = *(const v2f*)&ldsW[(d0 + m) * LDA + 4 * kk + khalf];
            acc = __builtin_amdgcn_wmma_f32_16x16x4_f32(
                false, a, false, b, (short)0, acc, false, false);
        }
        float bd = bias[d0 + m];
        #pragma unroll
        for (int g = 0; g < 8; ++g)
            Y[(size_t)ybase[g] + d0 + m] = acc[g] + bd;
    }
}

// ---------------- K2a: partial column sums of Y (16 row-chunks) ----------------
__global__ __launch_bounds__(256) void k_stats_partial(
    const float* __restrict__ Y, float* __restrict__ psum, float* __restrict__ psq)
{
    __shared__ float ls[4][64], ls2[4][64];
    int cLocal = threadIdx.x & 63;
    int rg     = threadIdx.x >> 6;          // 4 row groups
    int col    = blockIdx.x * 64 + cLocal;  // 25 col-blocks
    int chunk  = blockIdx.y;                // 16 chunks of 1200 rows
    int rend   = chunk * 1200 + 1200;
    float s = 0.f, s2 = 0.f;
    for (int r = chunk * 1200 + rg; r < rend; r += 4) {
        float y = Y[(size_t)r * COLS + col];
        s += y; s2 += y * y;
    }
    ls[rg][cLocal] = s; ls2[rg][cLocal] = s2;
    __syncthreads();
    if (rg == 0) {
        s  = ls[0][cLocal]  + ls[1][cLocal]  + ls[2][cLocal]  + ls[3][cLocal];
        s2 = ls2[0][cLocal] + ls2[1][cLocal] + ls2[2][cLocal] + ls2[3][cLocal];
        psum[chunk * COLS + col] = s;
        psq [chunk * COLS + col] = s2;
    }
}

// ---------------- K2b: finalize mean / inv-std ----------------
__global__ void k_stats_final(const float* __restrict__ psum, const float* __restrict__ psq,
                              float* __restrict__ mean, float* __restrict__ istd)
{
    int j = blockIdx.x * blockDim.x + threadIdx.x;
    if (j >= COLS) return;
    float s = 0.f, s2 = 0.f;
    #pragma unroll
    for (int c = 0; c < 16; ++c) { s += psum[c * COLS + j]; s2 += psq[c * COLS + j]; }
    float inv_n = 1.0f / (float)NT;
    float mu  = s * inv_n;
    float var = s2 * inv_n - mu * mu;
    mean[j] = mu;
    istd[j] = rsqrtf(var + BN_EPS);
}

// ---------------- K3: gather(shift_out) + BN + residual + ReLU, (N,D,T,V) layout ----------------
__global__ __launch_bounds__(256) void k_out(
    const float* __restrict__ Y, const float* __restrict__ x0,
    const int* __restrict__ shift_out, const float* __restrict__ gamma,
    const float* __restrict__ beta, const float* __restrict__ mean,
    const float* __restrict__ istd, float* __restrict__ out)
{
    int o  = blockIdx.x * 256 + threadIdx.x;     // 30,720,000 exactly
    int v  = o % VN;
    int t  = (o / VN) % TDIM;
    int d  = (o / (VN * TDIM)) & 63;
    int n  = o / (VN * TDIM * 64);
    int nt = n * TDIM + t;
    int j  = v * 64 + d;
    int so = shift_out[j];
    float y   = Y[(size_t)nt * COLS + so];
    float val = (y - mean[so]) * istd[so] * gamma[j] + beta[j] + x0[o];
    out[o] = fmaxf(val, 0.f);
}

extern "C" void kernel_launch(void* const* d_in, const int* in_sizes, int n_in,
                              void* d_out, int out_size, void* d_ws, size_t ws_size,
                              hipStream_t stream) {
    const float* x0        = (const float*)d_in[0];
    const float* fm        = (const float*)d_in[1];
    const float* W         = (const float*)d_in[2];
    const float* b         = (const float*)d_in[3];
    const float* gamma     = (const float*)d_in[4];
    const float* beta      = (const float*)d_in[5];
    const int*   shift_in  = (const int*)d_in[6];
    const int*   shift_out = (const int*)d_in[7];
    float* out = (float*)d_out;

    // Workspace layout (floats): Y | mask | psum(16x1600) | psq(16x1600) | mean | istd
    float* Y    = (float*)d_ws;
    float* mask = Y + (size_t)ROWS * 64;          // 30,720,000
    float* psum = mask + COLS;
    float* psq  = psum + 16 * COLS;
    float* mean = psq  + 16 * COLS;
    float* istd = mean + COLS;

    k_mask<<<(COLS + 255) / 256, 256, 0, stream>>>(fm, mask);
    k_gemm<<<ROWS / 128, 256, 0, stream>>>(x0, W, b, shift_in, mask, Y);
    k_stats_partial<<<dim3(COLS / 64, 16), 256, 0, stream>>>(Y, psum, psq);
    k_stats_final<<<(COLS + 255) / 256, 256, 0, stream>>>(psum, psq, mean, istd);
    k_out<<<(NB * 64 * TDIM * VN) / 256, 256, 0, stream>>>(Y, x0, shift_out, gamma, beta,
                                                           mean, istd, out);
}